// DistanceLossFast_69277822484656
// MI455X (gfx1250) — compile-verified
//
#include <hip/hip_runtime.h>
#include <cstdint>
#include <cstddef>

// Problem constants (fixed by the reference's setup_inputs)
#define BB  64
#define LL  128
#define HH  512
#define WW  512
constexpr int   HW        = HH * WW;
constexpr float DELTA_T   = 2.0e-4f;
constexpr int   MAX_STEPS = 200;
constexpr float MAX_RHO   = 0.5f * (float)WW;   // min(H,W)/2 = 256

// jnp.gradient along width (axis=2): central diff interior, one-sided edges
__device__ __forceinline__ float gradX(const float* __restrict__ img, int y, int x) {
    const float* row = img + (size_t)y * WW;
    if (x == 0)       return row[1] - row[0];
    if (x == WW - 1)  return row[WW - 1] - row[WW - 2];
    return 0.5f * (row[x + 1] - row[x - 1]);
}
// jnp.gradient along height (axis=1)
__device__ __forceinline__ float gradY(const float* __restrict__ img, int y, int x) {
    if (y == 0)       return img[(size_t)WW + x] - img[x];
    if (y == HH - 1)  return img[(size_t)(HH - 1) * WW + x] - img[(size_t)(HH - 2) * WW + x];
    return 0.5f * (img[(size_t)(y + 1) * WW + x] - img[(size_t)(y - 1) * WW + x]);
}

// One workgroup per batch; one ray per thread (128 threads = 4 wave32 waves).
extern "C" __global__ void __launch_bounds__(LL)
darnet_evolve(const float* __restrict__ rho_init,
              const float* __restrict__ rho_target,
              const float* __restrict__ origin,
              const float* __restrict__ beta,
              const float* __restrict__ data,
              const float* __restrict__ kappa,
              const float* __restrict__ theta,
              const float* __restrict__ delta_theta,
              float* __restrict__ out,        // [0]=diff, [1..]=cx, cy, rho
              float* __restrict__ partial)    // [BB] per-batch |rho-rt| sums
{
    __shared__ float rbuf[2][LL];   // double-buffered rho ring
    __shared__ float red[LL];       // reduction scratch

    const int b = blockIdx.x;
    const int t = threadIdx.x;      // ray index 0..127

    // ---- Stage initial rho row into LDS via gfx1250 async-LDS load ----
#if defined(__gfx1250__)
    {
        unsigned lds_off = (unsigned)(uintptr_t)(&rbuf[0][t]);   // group-relative LDS offset
        const float* gsrc = rho_init + (size_t)b * LL + t;       // per-lane global addr
        asm volatile("global_load_async_to_lds_b32 %0, %1, off"
                     :: "v"(lds_off), "v"(gsrc) : "memory");
    }
#else
    rbuf[0][t] = rho_init[(size_t)b * LL + t];
#endif

    // Loop-invariant per-ray state (overlaps with async staging above)
    const float th = theta[(size_t)b * LL + t];
    float st, ct;
    __sincosf(th, &st, &ct);        // hw v_sin/v_cos; theta in [0,2pi)
    const float ox = origin[2 * b + 0];
    const float oy = origin[2 * b + 1];
    const float dt = delta_theta[0];
    const float inv2 = 1.0f / (dt * dt);
    const float inv4 = inv2 * inv2;

    const float* __restrict__ D  = data  + (size_t)b * HW;
    const float* __restrict__ Bt = beta  + (size_t)b * HW;
    const float* __restrict__ K  = kappa + (size_t)b * HW;

#if defined(__gfx1250__)
    asm volatile("s_wait_asynccnt 0x0" ::: "memory");
#endif
    __syncthreads();

    float rho = rbuf[0][t];
    int cur = 0;

    // Cold-start prefetch of first step's bilinear corner cachelines
    {
        float x = fminf(fmaxf(ox + rho * ct, 0.0f), (float)(WW - 1));
        float y = fminf(fmaxf(oy + rho * st, 0.0f), (float)(HH - 1));
        const int x0 = (int)floorf(x), y0 = (int)floorf(y);
        const int x1 = min(x0 + 1, WW - 1), y1 = min(y0 + 1, HH - 1);
        const size_t i00 = (size_t)y0 * WW + x0;
        const size_t i11 = (size_t)y1 * WW + x1;
        __builtin_prefetch(D  + i00, 0, 3);   // global_prefetch_b8
        __builtin_prefetch(D  + i11, 0, 3);
        __builtin_prefetch(Bt + i00, 0, 3);
        __builtin_prefetch(Bt + i11, 0, 3);
        __builtin_prefetch(K  + i00, 0, 3);
        __builtin_prefetch(K  + i11, 0, 3);
    }

    for (int s = 0; s < MAX_STEPS; ++s) {
        // Circular 5-tap neighborhood from LDS ring
        const float rm1 = rbuf[cur][(t + LL - 1) & (LL - 1)];
        const float rp1 = rbuf[cur][(t + 1)      & (LL - 1)];
        const float rm2 = rbuf[cur][(t + LL - 2) & (LL - 1)];
        const float rp2 = rbuf[cur][(t + 2)      & (LL - 1)];

        // Contour point, clipped to image
        float x = ox + rho * ct;
        float y = oy + rho * st;
        x = fminf(fmaxf(x, 0.0f), (float)(WW - 1));
        y = fminf(fmaxf(y, 0.0f), (float)(HH - 1));

        const float x0f = floorf(x);
        const float y0f = floorf(y);
        const float wx = x - x0f;
        const float wy = y - y0f;
        const int x0 = (int)x0f;
        const int y0 = (int)y0f;
        const int x1 = min(x0 + 1, WW - 1);
        const int y1 = min(y0 + 1, HH - 1);

        const float w00 = (1.0f - wx) * (1.0f - wy);
        const float w01 = wx * (1.0f - wy);
        const float w10 = (1.0f - wx) * wy;
        const float w11 = wx * wy;

        const size_t i00 = (size_t)y0 * WW + x0;
        const size_t i01 = (size_t)y0 * WW + x1;
        const size_t i10 = (size_t)y1 * WW + x0;
        const size_t i11 = (size_t)y1 * WW + x1;

        // Bilinear samples of on-the-fly image gradients (L2/WGP$-resident)
        const float gx = w00 * gradX(D, y0, x0) + w01 * gradX(D, y0, x1)
                       + w10 * gradX(D, y1, x0) + w11 * gradX(D, y1, x1);
        const float gy = w00 * gradY(D, y0, x0) + w01 * gradY(D, y0, x1)
                       + w10 * gradY(D, y1, x0) + w11 * gradY(D, y1, x1);
        const float bb = w00 * Bt[i00] + w01 * Bt[i01] + w10 * Bt[i10] + w11 * Bt[i11];
        const float kk = w00 * K[i00]  + w01 * K[i01]  + w10 * K[i10]  + w11 * K[i11];

        const float dD = gx * ct + gy * st;
        const float d2 = (rp1 - 2.0f * rho + rm1) * inv2;
        const float d4 = (rp2 - 4.0f * rp1 + 6.0f * rho - 4.0f * rm1 + rm2) * inv4;

        rho = rho - DELTA_T * (dD - bb * d2 + kk * d4);
        rho = fminf(fmaxf(rho, 1.0f), MAX_RHO);

        cur ^= 1;
        rbuf[cur][t] = rho;
        __syncthreads();   // single barrier per step (double buffering)
    }

    // ---- Outputs ----
    const size_t idx = (size_t)b * LL + t;
    out[1 + idx]                   = ox + rho * ct;   // contour_x
    out[1 + (size_t)BB*LL + idx]   = oy + rho * st;   // contour_y
    out[1 + 2*(size_t)BB*LL + idx] = rho;             // rho

    // ---- Deterministic per-batch |rho - target| partial sum ----
    red[t] = fabsf(rho - rho_target[idx]);
    __syncthreads();
    #pragma unroll
    for (int s = LL / 2; s > 0; s >>= 1) {
        if (t < s) red[t] += red[t + s];
        __syncthreads();
    }
    if (t == 0) partial[b] = red[0];
}

// Ordered finish: sum 64 per-batch partials -> mean abs diff (deterministic)
extern "C" __global__ void darnet_finish(const float* __restrict__ partial,
                                         float* __restrict__ out)
{
    if (threadIdx.x == 0 && blockIdx.x == 0) {
        float s = 0.0f;
        #pragma unroll
        for (int i = 0; i < BB; ++i) s += partial[i];
        out[0] = s * (1.0f / (float)(BB * LL));
    }
}

extern "C" void kernel_launch(void* const* d_in, const int* in_sizes, int n_in,
                              void* d_out, int out_size, void* d_ws, size_t ws_size,
                              hipStream_t stream)
{
    (void)in_sizes; (void)n_in; (void)out_size; (void)ws_size;
    const float* rho_init    = (const float*)d_in[0];
    const float* rho_target  = (const float*)d_in[1];
    const float* origin      = (const float*)d_in[2];
    const float* beta        = (const float*)d_in[3];
    const float* data        = (const float*)d_in[4];
    const float* kappa       = (const float*)d_in[5];
    const float* theta       = (const float*)d_in[6];
    const float* delta_theta = (const float*)d_in[7];
    float* out     = (float*)d_out;
    float* partial = (float*)d_ws;   // 64 floats of scratch

    darnet_evolve<<<dim3(BB), dim3(LL), 0, stream>>>(
        rho_init, rho_target, origin, beta, data, kappa, theta, delta_theta,
        out, partial);
    darnet_finish<<<dim3(1), dim3(32), 0, stream>>>(partial, out);
}